// GaussianNoiseInjector_15848429322599
// MI455X (gfx1250) — compile-verified
//
#include <hip/hip_runtime.h>
#include <hip/hip_bf16.h>
#include <math.h>
#include <stdint.h>

// ---------------------------------------------------------------------------
// GaussianNoiseInjector for MI455X (gfx1250, wave32)
//   pass 1: per-(b,c) row mean via V_WMMA_F32_16X16X4_F32 (ones-vector matmul),
//           tile staged to LDS with GLOBAL_LOAD_ASYNC_TO_LDS_B128 (ASYNCcnt)
//   pass 2: out = x + mask[b] * noise_std[b,c] * noise   (float4, bandwidth-bound)
// ---------------------------------------------------------------------------

typedef __attribute__((ext_vector_type(2))) float v2f;
typedef __attribute__((ext_vector_type(8))) float v8f;

#define B_DIM 256
#define C_DIM 64
#define T_DIM 4096
#define ROWS  (B_DIM * C_DIM)        // 16384 (b,c) rows
#define LSTRIDE 68                   // LDS row stride (floats): 272B = 16B-aligned, bank-conflict-free
#define INV_T (1.0f / 4096.0f)
#define NOISE_POW 0.1f               // 10^(-SNR_DB/10), SNR_DB = 10

// ---------------------------------------------------------------------------
// Pass 1: 16 rows per block, 256 threads = 8 wave32s.
// Each 64-column chunk of the 16x4096 row tile is DMA'd straight into LDS via
// the async global->LDS path (no VGPR bounce), then the 8 waves split the 16
// K-steps (K=4 each) and accumulate row sums on the matrix pipe:
//   D = A(16x4) * ones(4x16) + C.
// ---------------------------------------------------------------------------
__global__ __launch_bounds__(256) void row_std_wmma(const float* __restrict__ x,
                                                    float* __restrict__ stdv) {
  __shared__ __align__(16) float tile[16 * LSTRIDE];
  __shared__ float red[8 * 16];

  const int tid  = threadIdx.x;
  const int rowL = tid >> 4;          // 0..15: which of the 16 rows this thread loads
  const int jj   = tid & 15;          // 16 threads cover 64 contiguous floats of a row
  const int lane = tid & 31;
  const int wave = tid >> 5;          // 0..7
  const int m    = lane & 15;         // A-fragment row (M) for this lane
  const int koff = (lane >= 16) ? 2 : 0; // ISA A layout: lanes 0-15 hold K=0,1; 16-31 hold K=2,3
  const int rowBase = blockIdx.x << 4;

  const float* rowPtr = x + (size_t)(rowBase + rowL) * T_DIM;

  // LDS byte address of this thread's 16-byte staging slot.
  // Generic->LDS mapping: the low 32 bits of the address give the LDS offset.
  const uint32_t lds_slot =
      (uint32_t)(uintptr_t)&tile[rowL * LSTRIDE + (jj << 2)];

  v8f c = {};
  v2f ones_b;
  ones_b.x = 1.0f;
  ones_b.y = 1.0f;

  for (int t0 = 0; t0 < T_DIM; t0 += 64) {
    // Async DMA: 16 bytes per lane, memory -> LDS directly (ASYNCcnt).
    const uint64_t gaddr = (uint64_t)(uintptr_t)(rowPtr + t0 + (jj << 2));
    asm volatile("global_load_async_to_lds_b128 %0, %1, off"
                 :: "v"(lds_slot), "v"(gaddr)
                 : "memory");
    // Wait for our own async transfers to land in LDS, then make them
    // visible block-wide.
    asm volatile("s_wait_asynccnt 0x0" ::: "memory");
    __syncthreads();

    // 16 K-steps in this chunk; wave w takes steps 2w and 2w+1.
#pragma unroll
    for (int j = 0; j < 2; ++j) {
      const int kk = (((wave << 1) + j) << 2) + koff;
      v2f a;
      a.x = tile[m * LSTRIDE + kk];
      a.y = tile[m * LSTRIDE + kk + 1];
      // D[m,n] = C[m,n] + sum_k A[m,k] * 1  -> row sums, replicated over N
      c = __builtin_amdgcn_wmma_f32_16x16x4_f32(
          /*neg_a=*/false, a, /*neg_b=*/false, ones_b,
          /*c_mod=*/(short)0, c, /*reuse_a=*/false, /*reuse_b=*/false);
    }
    __syncthreads();
  }

  // C/D 16x16 f32 layout: VGPR r, lanes 0-15 -> M=r; lanes 16-31 -> M=8+r.
  // Lane 0 (N=0) holds rows 0..7, lane 16 holds rows 8..15.
  if (lane == 0) {
#pragma unroll
    for (int r = 0; r < 8; ++r) red[wave * 16 + r] = c[r];
  }
  if (lane == 16) {
#pragma unroll
    for (int r = 0; r < 8; ++r) red[wave * 16 + 8 + r] = c[r];
  }
  __syncthreads();

  // Combine the 8 waves' partial sums and emit noise_std for 16 rows.
  if (tid < 16) {
    float s = 0.0f;
#pragma unroll
    for (int w = 0; w < 8; ++w) s += red[w * 16 + tid];
    const float p_sig = fabsf(s * INV_T);
    stdv[rowBase + tid] = sqrtf(p_sig * NOISE_POW);
  }
}

// ---------------------------------------------------------------------------
// Pass 2: out = x + (mask[b] ? noise * std[b,c] : 0), one float4 per thread.
// 768 MB of b128 traffic -> this kernel decides overall runtime (~33 us at
// 23.3 TB/s for this pass alone).
// ---------------------------------------------------------------------------
__global__ __launch_bounds__(256) void add_gated_noise(const float* __restrict__ x,
                                                       const float* __restrict__ nz,
                                                       const unsigned char* __restrict__ mask,
                                                       const float* __restrict__ stdv,
                                                       float* __restrict__ out) {
  const size_t i4 = (size_t)blockIdx.x * 256 + threadIdx.x; // float4 index
  const size_t i  = i4 << 2;                                // element index
  const int row = (int)(i >> 12);                           // / T (4096): b*C + c
  const int b   = row >> 6;                                 // / C (64)
  const float s = mask[b] ? stdv[row] : 0.0f;

  const float4 xv = ((const float4*)x)[i4];
  const float4 nv = ((const float4*)nz)[i4];
  float4 o;
  o.x = fmaf(nv.x, s, xv.x);
  o.y = fmaf(nv.y, s, xv.y);
  o.z = fmaf(nv.z, s, xv.z);
  o.w = fmaf(nv.w, s, xv.w);
  ((float4*)out)[i4] = o;
}

// ---------------------------------------------------------------------------
extern "C" void kernel_launch(void* const* d_in, const int* in_sizes, int n_in,
                              void* d_out, int out_size, void* d_ws, size_t ws_size,
                              hipStream_t stream) {
  (void)in_sizes; (void)n_in; (void)out_size; (void)ws_size;

  const float*         x    = (const float*)d_in[0];          // [B,1,C,T] f32
  const float*         nz   = (const float*)d_in[1];          // [B,C,T]   f32
  const unsigned char* mask = (const unsigned char*)d_in[2];  // [B] bool (1 byte/elem)
  float*               out  = (float*)d_out;                  // [B,1,C,T] f32
  float*               stdv = (float*)d_ws;                   // 16384 f32 scratch (64 KB)

  // Pass 1: 1024 blocks x 256 threads, 16 rows each.
  row_std_wmma<<<ROWS / 16, 256, 0, stream>>>(x, stdv);

  // Pass 2: 67,108,864 elements / 4 per thread / 256 per block = 65536 blocks.
  const unsigned n_blocks = (unsigned)(((size_t)B_DIM * C_DIM * T_DIM / 4) / 256);
  add_gated_noise<<<n_blocks, 256, 0, stream>>>(x, nz, mask, stdv, out);
}